// Fusion_Layer_45835890983493
// MI455X (gfx1250) — compile-verified
//
#include <hip/hip_runtime.h>

typedef __attribute__((ext_vector_type(2))) float v2f;
typedef __attribute__((ext_vector_type(4))) float v4f;
typedef __attribute__((ext_vector_type(8))) float v8f;

#define B_ 8
#define N_ 16384
#define C_ 64
#define S_ 5
#define OUT_OFF ((size_t)B_ * N_ * 3)

// ---------------------------------------------------------------------------
// Kernel 1: farthest point sampling (5 samples, one workgroup per batch).
// Distances live in registers (16 points per thread); LDS only for argmax.
// ---------------------------------------------------------------------------
__global__ __launch_bounds__(1024) void fps_kernel(const float* __restrict__ bg_xyz,
                                                   int* __restrict__ idx_out) {
  const int b   = blockIdx.x;
  const int tid = threadIdx.x;
  const float* xyz = bg_xyz + (size_t)b * N_ * 3;

  constexpr int PPT = N_ / 1024;  // 16 points per thread, strided
  float px[PPT], py[PPT], pz[PPT], dist[PPT];
#pragma unroll
  for (int j = 0; j < PPT; ++j) {
    int i = tid + j * 1024;
    px[j] = xyz[3 * i + 0];
    py[j] = xyz[3 * i + 1];
    pz[j] = xyz[3 * i + 2];
    dist[j] = 1e10f;
  }

  __shared__ float sval[1024];
  __shared__ int   sidx[1024];
  __shared__ float lastp[3];

  int last = 0;
  for (int it = 0; it < S_; ++it) {
    if (tid == 0) {
      idx_out[b * S_ + it] = last;   // reference scan emits the *incoming* index
      lastp[0] = xyz[3 * last + 0];
      lastp[1] = xyz[3 * last + 1];
      lastp[2] = xyz[3 * last + 2];
    }
    __syncthreads();
    const float lx = lastp[0], ly = lastp[1], lz = lastp[2];

    float bv = -1.0f;
    int   bi = 0x7fffffff;
#pragma unroll
    for (int j = 0; j < PPT; ++j) {
      float dx = px[j] - lx, dy = py[j] - ly, dz = pz[j] - lz;
      float d  = dx * dx + dy * dy + dz * dz;
      dist[j]  = fminf(dist[j], d);
      if (dist[j] > bv) { bv = dist[j]; bi = tid + j * 1024; }  // strict > keeps smallest idx in-thread
    }
    sval[tid] = bv;
    sidx[tid] = bi;
    __syncthreads();
    for (int stride = 512; stride > 0; stride >>= 1) {
      if (tid < stride) {
        float ov = sval[tid + stride];
        int   oi = sidx[tid + stride];
        if (ov > sval[tid] || (ov == sval[tid] && oi < sidx[tid])) {
          sval[tid] = ov;
          sidx[tid] = oi;
        }
      }
      __syncthreads();
    }
    last = sidx[0];
    __syncthreads();
  }
}

// ---------------------------------------------------------------------------
// Kernel 2: gather sampled bg features + precompute cvec[b][s] = W1b @ sbg[b][s]
// ---------------------------------------------------------------------------
__global__ __launch_bounds__(C_) void gather_kernel(const float* __restrict__ bg_features,
                                                    const float* __restrict__ w1,
                                                    const int* __restrict__ idx,
                                                    float* __restrict__ sbg,
                                                    float* __restrict__ cvec) {
  const int s = blockIdx.x;   // 0..S-1
  const int b = blockIdx.y;
  const int o = threadIdx.x;  // 0..63
  __shared__ float sv[C_];

  const int i = idx[b * S_ + s];
  sv[o] = bg_features[((size_t)b * C_ + o) * N_ + i];
  __syncthreads();

  float acc = 0.0f;
#pragma unroll 8
  for (int c = 0; c < C_; ++c) acc += w1[o * (2 * C_) + C_ + c] * sv[c];

  sbg [(b * S_ + s) * C_ + o] = sv[o];
  cvec[(b * S_ + s) * C_ + o] = acc;
}

// ---------------------------------------------------------------------------
// Kernel 3: fused MLP. 8 waves/block, each wave owns a 16-column strip and
// the full 64-channel output, using V_WMMA_F32_16X16X4_F32 (fp32 matrix op).
// ---------------------------------------------------------------------------
__global__ __launch_bounds__(256) void fused_mlp_kernel(
    const float* __restrict__ fg_features,
    const float* __restrict__ w1, const float* __restrict__ w2,
    const float* __restrict__ g1, const float* __restrict__ b1,
    const float* __restrict__ m1, const float* __restrict__ v1,
    const float* __restrict__ g2, const float* __restrict__ b2,
    const float* __restrict__ m2, const float* __restrict__ v2,
    const float* __restrict__ sbg, const float* __restrict__ cvec,
    float* __restrict__ out) {
  __shared__ float ls1[C_], lt1[C_], ls2[C_], lt2[C_];
  __shared__ float lsbg[S_ * C_], lcv[S_ * C_];
  __shared__ float hbuf[8][16 * C_];  // per-wave h staging, layout [col][chan]

  const int b   = blockIdx.y;
  const int tid = threadIdx.x;

  if (tid < C_) {
    float s1 = g1[tid] * rsqrtf(v1[tid] + 1e-5f);
    ls1[tid] = s1;
    lt1[tid] = b1[tid] - m1[tid] * s1;
    float s2 = g2[tid] * rsqrtf(v2[tid] + 1e-5f);
    ls2[tid] = s2;
    lt2[tid] = b2[tid] - m2[tid] * s2;
  }
  for (int i = tid; i < S_ * C_; i += 256) {
    lsbg[i] = sbg [b * S_ * C_ + i];
    lcv[i]  = cvec[b * S_ * C_ + i];
  }
  __syncthreads();

  const int wave = tid >> 5, lane = tid & 31;
  const int half = lane >> 4, lr = lane & 15;
  const int koff = 2 * half;
  const int n    = blockIdx.x * 128 + wave * 16 + lr;  // this lane's column
  const float* fg = fg_features + (size_t)b * C_ * N_;
  float* hb = &hbuf[wave][0];

  const v8f vzero = {0.f, 0.f, 0.f, 0.f, 0.f, 0.f, 0.f, 0.f};
  v8f acc1[4], acc2[4], oacc[4];
#pragma unroll
  for (int m = 0; m < 4; ++m) { acc1[m] = vzero; acc2[m] = vzero; oacc[m] = vzero; }

  // -------- Layer 1: acc1 = W1a @ fg_tile ; acc2 = W1b @ fg_tile -----------
#pragma unroll 4
  for (int step = 0; step < 16; ++step) {
    const int k0 = step * 4 + koff;
    v2f bv;
    bv.x = fg[(size_t)(k0 + 0) * N_ + n];
    bv.y = fg[(size_t)(k0 + 1) * N_ + n];
#pragma unroll
    for (int m = 0; m < 4; ++m) {
      const float* wr = w1 + (size_t)(m * 16 + lr) * (2 * C_) + k0;
      v2f a1 = *(const v2f*)(wr);
      v2f a2 = *(const v2f*)(wr + C_);
      acc1[m] = __builtin_amdgcn_wmma_f32_16x16x4_f32(false, a1, false, bv, (short)0,
                                                      acc1[m], false, false);
      acc2[m] = __builtin_amdgcn_wmma_f32_16x16x4_f32(false, a2, false, bv, (short)0,
                                                      acc2[m], false, false);
    }
  }

  // -------- Loop over the 6 "s" slices -------------------------------------
  for (int ss = 0; ss <= S_; ++ss) {
    // h = relu(s1 * (acc1 + add) + t1), staged to LDS transposed ([col][chan])
#pragma unroll
    for (int m = 0; m < 4; ++m) {
      const int obase = m * 16 + 8 * half;  // first channel this lane owns in block m
      v4f s1a = *(const v4f*)&ls1[obase], s1b = *(const v4f*)&ls1[obase + 4];
      v4f t1a = *(const v4f*)&lt1[obase], t1b = *(const v4f*)&lt1[obase + 4];
      v4f ha, hc;
      if (ss == 0) {
#pragma unroll
        for (int v = 0; v < 4; ++v) {
          ha[v] = fmaxf((acc1[m][v]     + acc2[m][v])     * s1a[v] + t1a[v], 0.f);
          hc[v] = fmaxf((acc1[m][v + 4] + acc2[m][v + 4]) * s1b[v] + t1b[v], 0.f);
        }
      } else {
        v4f ca = *(const v4f*)&lcv[(ss - 1) * C_ + obase];
        v4f cb = *(const v4f*)&lcv[(ss - 1) * C_ + obase + 4];
#pragma unroll
        for (int v = 0; v < 4; ++v) {
          ha[v] = fmaxf((acc1[m][v]     + ca[v]) * s1a[v] + t1a[v], 0.f);
          hc[v] = fmaxf((acc1[m][v + 4] + cb[v]) * s1b[v] + t1b[v], 0.f);
        }
      }
      *(v4f*)&hb[lr * C_ + obase]     = ha;  // ds_store_b128
      *(v4f*)&hb[lr * C_ + obase + 4] = hc;
    }
    // per-wave private LDS region; same-wave LDS ops are in-order -> no barrier

    // z = W2 @ h
    v8f zac[4];
#pragma unroll
    for (int m = 0; m < 4; ++m) zac[m] = vzero;
#pragma unroll 4
    for (int step = 0; step < 16; ++step) {
      const int k0 = step * 4 + koff;
      v2f bh = *(const v2f*)&hb[lr * C_ + k0];  // ds_load_b64
#pragma unroll
      for (int m = 0; m < 4; ++m) {
        v2f a = *(const v2f*)(w2 + (size_t)(m * 16 + lr) * C_ + k0);
        zac[m] = __builtin_amdgcn_wmma_f32_16x16x4_f32(false, a, false, bh, (short)0,
                                                       zac[m], false, false);
      }
    }

    // wf = relu(s2*z + t2);  oacc += wf * fused_origin[ss]
#pragma unroll
    for (int m = 0; m < 4; ++m) {
      const int obase = m * 16 + 8 * half;
      v4f s2a = *(const v4f*)&ls2[obase], s2b = *(const v4f*)&ls2[obase + 4];
      v4f t2a = *(const v4f*)&lt2[obase], t2b = *(const v4f*)&lt2[obase + 4];
      if (ss == 0) {
#pragma unroll
        for (int v = 0; v < 8; ++v) {
          float sc = (v < 4) ? s2a[v] : s2b[v - 4];
          float tc = (v < 4) ? t2a[v] : t2b[v - 4];
          float wf = fmaxf(zac[m][v] * sc + tc, 0.f);
          float og = fg[(size_t)(obase + v) * N_ + n];
          oacc[m][v] += wf * og;
        }
      } else {
        v4f oa = *(const v4f*)&lsbg[(ss - 1) * C_ + obase];
        v4f ob = *(const v4f*)&lsbg[(ss - 1) * C_ + obase + 4];
#pragma unroll
        for (int v = 0; v < 8; ++v) {
          float sc = (v < 4) ? s2a[v] : s2b[v - 4];
          float tc = (v < 4) ? t2a[v] : t2b[v - 4];
          float wf = fmaxf(zac[m][v] * sc + tc, 0.f);
          float og = (v < 4) ? oa[v] : ob[v - 4];
          oacc[m][v] += wf * og;
        }
      }
    }
  }

  // -------- Write final_feats: out[b][o][n] --------------------------------
  float* outf = out + OUT_OFF + (size_t)b * C_ * N_;
#pragma unroll
  for (int m = 0; m < 4; ++m) {
    const int obase = m * 16 + 8 * half;
#pragma unroll
    for (int v = 0; v < 8; ++v) {
      outf[(size_t)(obase + v) * N_ + n] = oacc[m][v];
    }
  }
}

// ---------------------------------------------------------------------------
extern "C" void kernel_launch(void* const* d_in, const int* in_sizes, int n_in,
                              void* d_out, int out_size, void* d_ws, size_t ws_size,
                              hipStream_t stream) {
  const float* fg_xyz      = (const float*)d_in[0];
  const float* fg_features = (const float*)d_in[1];
  const float* bg_xyz      = (const float*)d_in[2];
  const float* bg_features = (const float*)d_in[3];
  const float* w1          = (const float*)d_in[4];
  const float* g1          = (const float*)d_in[5];
  const float* b1          = (const float*)d_in[6];
  const float* m1          = (const float*)d_in[7];
  const float* v1          = (const float*)d_in[8];
  const float* w2          = (const float*)d_in[9];
  const float* g2          = (const float*)d_in[10];
  const float* b2          = (const float*)d_in[11];
  const float* m2          = (const float*)d_in[12];
  const float* v2          = (const float*)d_in[13];

  int*   idx  = (int*)d_ws;
  float* sbg  = (float*)((char*)d_ws + 256);
  float* cvec = sbg + B_ * S_ * C_;
  float* out  = (float*)d_out;

  // Output tuple element 0: fg_xyz passthrough.
  hipMemcpyAsync(d_out, d_in[0], OUT_OFF * sizeof(float),
                 hipMemcpyDeviceToDevice, stream);

  fps_kernel<<<B_, 1024, 0, stream>>>(bg_xyz, idx);
  gather_kernel<<<dim3(S_, B_), C_, 0, stream>>>(bg_features, w1, idx, sbg, cvec);
  fused_mlp_kernel<<<dim3(N_ / 128, B_), 256, 0, stream>>>(
      fg_features, w1, w2, g1, b1, m1, v1, g2, b2, m2, v2, sbg, cvec, out);
}